// TGNLinkPredictor_34050500723061
// MI455X (gfx1250) — compile-verified
//
#include <hip/hip_runtime.h>
#include <math.h>

// ---------------- problem constants (from reference) ----------------
#define DDIM   100      // D
#define CCH    50       // per-head channels
#define EDIM   101      // D + MSG_DIM
#define NN_CAP 1000000  // num_nodes upper bound (assoc table capacity)

typedef float v2f __attribute__((ext_vector_type(2)));
typedef float v8f __attribute__((ext_vector_type(8)));

// order-preserving float<->uint encoding for atomicMax-based segment max
__device__ __forceinline__ unsigned f2ord(float f) {
  unsigned u = __float_as_uint(f);
  return (u & 0x80000000u) ? ~u : (u | 0x80000000u);
}
__device__ __forceinline__ float ord2f(unsigned u) {
  return (u & 0x80000000u) ? __uint_as_float(u & 0x7FFFFFFFu) : __uint_as_float(~u);
}

// ---------------- kernel 1: workspace init ----------------
__global__ void k_init(int* assoc, unsigned* amax, float* aden,
                       const int* __restrict__ num_nodes_p, int N) {
  int nn = *num_nodes_p;
  if (nn > NN_CAP) nn = NN_CAP;
  int total = (nn > 2 * N) ? nn : 2 * N;
  int stride = gridDim.x * blockDim.x;
  for (int i = blockIdx.x * blockDim.x + threadIdx.x; i < total; i += stride) {
    if (i < nn)     assoc[i] = 0;
    if (i < 2 * N) { amax[i] = 0u; aden[i] = 0.0f; }
  }
}

// ---------------- kernel 2: assoc scatter (global id -> local row) ----------------
__global__ void k_assoc(int* assoc, const int* __restrict__ n_id, int N) {
  int i = blockIdx.x * blockDim.x + threadIdx.x;
  if (i < N) assoc[n_id[i]] = i;
}

// ---------------- kernel 3: q/k/v/skip GEMMs via f32 WMMA ----------------
// Block = 256 threads = 8 waves. Each block: one of the 4 weight matrices,
// staged zero-padded into LDS ([112][100]); each wave owns one 16-row tile and
// iterates all 7 col tiles (7 accumulators held in VGPRs).
// A-frag (16x4 f32): lane<16 -> (M=lane, K=kk+j), lane>=16 -> (M=lane-16, K=kk+2+j)
// C/D: VGPR r -> (M = r + 8*(lane>=16), N = lane&15)
__global__ void __launch_bounds__(256)
k_qkvs(const float* __restrict__ z,
       const float* __restrict__ Wq, const float* __restrict__ bq,
       const float* __restrict__ Wk, const float* __restrict__ bk,
       const float* __restrict__ Wv, const float* __restrict__ bv,
       const float* __restrict__ Ws, const float* __restrict__ bs,
       float* __restrict__ qo, float* __restrict__ ko,
       float* __restrict__ vo, float* __restrict__ zout, int N) {
  __shared__ float Wlds[112 * DDIM];   // [col][kd], cols 100..111 zero
  __shared__ float blds[112];

  int tid  = threadIdx.x;
  int mat  = blockIdx.x & 3;
  int rt   = (blockIdx.x >> 2) * 8 + (tid >> 5);
  int lane = tid & 31;
  int nrt  = (N + 15) >> 4;

  const float* W   = (mat == 0) ? Wq : (mat == 1) ? Wk : (mat == 2) ? Wv : Ws;
  const float* bia = (mat == 0) ? bq : (mat == 1) ? bk : (mat == 2) ? bv : bs;
  float*       out = (mat == 0) ? qo : (mat == 1) ? ko : (mat == 2) ? vo : zout;

  // cooperative stage: layout col*100+kd is linear in W for col<100
  for (int i = tid; i < 112 * DDIM; i += 256)
    Wlds[i] = (i < DDIM * DDIM) ? W[i] : 0.0f;
  for (int i = tid; i < 112; i += 256)
    blds[i] = (i < DDIM) ? bia[i] : 0.0f;
  __syncthreads();

  if (rt >= nrt) return;  // wave-uniform exit (after barrier): EXEC all-ones for WMMA

  int  row   = rt * 16 + (lane & 15);
  bool rok   = row < N;
  int  rowc  = rok ? row : (N - 1);
  int  khalf = (lane >> 4) << 1;
  int  lcol  = lane & 15;
  const float* zrow = z + (size_t)rowc * DDIM + khalf;  // 8B-aligned (khalf even)

  v8f c[7] = {};
  for (int kk = 0; kk < DDIM; kk += 4) {  // 100 % 4 == 0
    v2f a = *(const v2f*)(zrow + kk);     // unconditional b64 load
    a[0] = rok ? a[0] : 0.0f;             // cndmask, no exec divergence
    a[1] = rok ? a[1] : 0.0f;
#pragma unroll
    for (int ct = 0; ct < 7; ++ct) {
      v2f b = *(const v2f*)(&Wlds[(ct * 16 + lcol) * DDIM + kk + khalf]);  // ds_load_b64
      c[ct] = __builtin_amdgcn_wmma_f32_16x16x4_f32(false, a, false, b,
                                                    (short)0, c[ct], false, false);
    }
  }
  int mbase = rt * 16 + ((lane >> 4) << 3);
#pragma unroll
  for (int ct = 0; ct < 7; ++ct) {
    int col = ct * 16 + lcol;
    if (col >= DDIM) continue;
    float bb = blds[col];
#pragma unroll
    for (int r = 0; r < 8; ++r) {
      int m = mbase + r;
      if (m < N) out[(size_t)m * DDIM + col] = c[ct][r] + bb;
    }
  }
}

// ---------------- kernel 4: fused time-encode + edge projection GEMM ----------------
// eproj[e, :] = concat(cos(rel_t[e]*time_w + time_b), msg[e]) @ We^T  (K 101 -> 104 pad)
// We staged zero-padded into LDS as [112][104].
__global__ void __launch_bounds__(256)
k_eproj(const float* __restrict__ last_update, const float* __restrict__ t,
        const float* __restrict__ msg,
        const float* __restrict__ time_w, const float* __restrict__ time_b,
        const float* __restrict__ We, const int* __restrict__ e_src,
        float* __restrict__ eproj, int E) {
  __shared__ float Welds[112 * 104];   // [col][kd], zero pad col>=100 or kd>=101
  __shared__ float twlds[DDIM];
  __shared__ float tblds[DDIM];

  int tid  = threadIdx.x;
  int wave = blockIdx.x * 8 + (tid >> 5);
  int lane = tid & 31;
  int net  = (E + 15) >> 4;

  for (int i = tid; i < 112 * 104; i += 256) {
    int col = i / 104, kd = i - col * 104;
    Welds[i] = (col < DDIM && kd < EDIM) ? We[col * EDIM + kd] : 0.0f;
  }
  for (int i = tid; i < DDIM; i += 256) {
    twlds[i] = time_w[i];
    tblds[i] = time_b[i];
  }
  __syncthreads();

  if (wave >= net) return;  // wave-uniform

  int  erow = wave * 16 + (lane & 15);
  bool rok  = erow < E;
  int  erc  = rok ? erow : (E - 1);
  float relt = last_update[e_src[erc]] - t[erc];  // unconditional (clamped)
  float m0   = msg[erc];
  int khalf = (lane >> 4) << 1;
  int lcol  = lane & 15;

  v8f c[7] = {};
  for (int kk = 0; kk < 104; kk += 4) {
    v2f a;
#pragma unroll
    for (int j = 0; j < 2; ++j) {
      int kd = kk + khalf + j;
      float av;
      if (kd < DDIM)       av = cosf(fmaf(relt, twlds[kd], tblds[kd]));
      else if (kd == DDIM) av = m0;
      else                 av = 0.0f;
      a[j] = rok ? av : 0.0f;
    }
#pragma unroll
    for (int ct = 0; ct < 7; ++ct) {
      v2f b = *(const v2f*)(&Welds[(ct * 16 + lcol) * 104 + kk + khalf]);  // ds_load_b64
      c[ct] = __builtin_amdgcn_wmma_f32_16x16x4_f32(false, a, false, b,
                                                    (short)0, c[ct], false, false);
    }
  }
  int mbase = wave * 16 + ((lane >> 4) << 3);
#pragma unroll
  for (int ct = 0; ct < 7; ++ct) {
    int col = ct * 16 + lcol;
    if (col >= DDIM) continue;
#pragma unroll
    for (int r = 0; r < 8; ++r) {
      int m = mbase + r;
      if (m < E) eproj[(size_t)m * DDIM + col] = c[ct][r];
    }
  }
}

// ---------------- kernel 5: alpha = q[dst]·(k[src]+eproj)/sqrt(C); segment max ----------------
__global__ void k_alpha(const float* __restrict__ q, const float* __restrict__ k,
                        const float* __restrict__ eproj,
                        const int* __restrict__ e_src, const int* __restrict__ e_dst,
                        float* __restrict__ alpha, unsigned* amax, int E) {
  int tid = blockIdx.x * blockDim.x + threadIdx.x;
  if (tid >= E * 2) return;
  int e = tid >> 1, h = tid & 1;
  int s = e_src[e], d = e_dst[e];
  const float* qp = q + (size_t)d * DDIM + h * CCH;
  const float* kp = k + (size_t)s * DDIM + h * CCH;
  const float* ep = eproj + (size_t)e * DDIM + h * CCH;
  float acc = 0.0f;
#pragma unroll 5
  for (int cc = 0; cc < CCH; ++cc) acc = fmaf(qp[cc], kp[cc] + ep[cc], acc);
  acc *= 0.14142135623730950488f;  // 1/sqrt(50)
  alpha[tid] = acc;
  atomicMax(&amax[d * 2 + h], f2ord(acc));
}

// ---------------- kernel 6: a = exp(alpha - max); segment sum ----------------
__global__ void k_expsum(const int* __restrict__ e_dst, const unsigned* __restrict__ amax,
                         float* __restrict__ alpha, float* aden, int E) {
  int tid = blockIdx.x * blockDim.x + threadIdx.x;
  if (tid >= E * 2) return;
  int e = tid >> 1, h = tid & 1;
  int d = e_dst[e];
  float a = expf(alpha[tid] - ord2f(amax[d * 2 + h]));
  alpha[tid] = a;
  atomicAdd(&aden[d * 2 + h], a);
}

// ---------------- kernel 7: weighted aggregation into zout (pre-init'd with skip) ----------------
__global__ void k_agg(const float* __restrict__ vmat, const float* __restrict__ eproj,
                      const float* __restrict__ alpha, const float* __restrict__ aden,
                      const int* __restrict__ e_src, const int* __restrict__ e_dst,
                      float* zout, int E) {
  long long tid = (long long)blockIdx.x * blockDim.x + threadIdx.x;
  long long tot = (long long)E * DDIM;
  if (tid >= tot) return;
  int e  = (int)(tid / DDIM);
  int cc = (int)(tid % DDIM);
  int h  = (cc >= CCH) ? 1 : 0;
  int d = e_dst[e], s = e_src[e];
  float w = alpha[e * 2 + h] / aden[d * 2 + h];
  atomicAdd(&zout[(size_t)d * DDIM + cc],
            (vmat[(size_t)s * DDIM + cc] + eproj[(size_t)e * DDIM + cc]) * w);
}

// ---------------- kernel 8: link predictor (two gathered GEMMs + ReLU) ----------------
__global__ void __launch_bounds__(256)
k_link(const float* __restrict__ zout,
       const float* __restrict__ Wsrc, const float* __restrict__ bsrc,
       const float* __restrict__ Wdst, const float* __restrict__ bdst,
       const int* __restrict__ assoc, const int* __restrict__ src,
       const int* __restrict__ dst, float* __restrict__ hbuf, int B) {
  int wave = (blockIdx.x * blockDim.x + threadIdx.x) >> 5;
  int lane = threadIdx.x & 31;
  int nbt  = (B + 15) >> 4;
  if (wave >= nbt) return;  // wave-uniform

  int  brow  = wave * 16 + (lane & 15);
  bool rok   = brow < B;
  int  browc = rok ? brow : (B - 1);
  int srow = assoc[src[browc]];   // unconditional (clamped) gathers
  int drow = assoc[dst[browc]];
  int khalf = (lane >> 4) << 1;
  int lcol  = lane & 15;
  const float* zs = zout + (size_t)srow * DDIM + khalf;
  const float* zd = zout + (size_t)drow * DDIM + khalf;

  v8f c[7] = {};
  for (int kk = 0; kk < DDIM; kk += 4) {
    v2f as = *(const v2f*)(zs + kk);
    v2f ad = *(const v2f*)(zd + kk);
    as[0] = rok ? as[0] : 0.0f;  as[1] = rok ? as[1] : 0.0f;
    ad[0] = rok ? ad[0] : 0.0f;  ad[1] = rok ? ad[1] : 0.0f;
#pragma unroll
    for (int ct = 0; ct < 7; ++ct) {
      int col  = ct * 16 + lcol;
      int colc = (col < DDIM) ? col : (DDIM - 1);
      v2f bsv = *(const v2f*)(Wsrc + (size_t)colc * DDIM + kk + khalf);
      v2f bdv = *(const v2f*)(Wdst + (size_t)colc * DDIM + kk + khalf);
      if (col >= DDIM) { bsv[0] = bsv[1] = bdv[0] = bdv[1] = 0.0f; }
      c[ct] = __builtin_amdgcn_wmma_f32_16x16x4_f32(false, as, false, bsv,
                                                    (short)0, c[ct], false, false);
      c[ct] = __builtin_amdgcn_wmma_f32_16x16x4_f32(false, ad, false, bdv,
                                                    (short)0, c[ct], false, false);
    }
  }
  int mbase = wave * 16 + ((lane >> 4) << 3);
#pragma unroll
  for (int ct = 0; ct < 7; ++ct) {
    int col = ct * 16 + lcol;
    if (col >= DDIM) continue;
    float bb = bsrc[col] + bdst[col];
#pragma unroll
    for (int r = 0; r < 8; ++r) {
      int m = mbase + r;
      if (m < B) hbuf[(size_t)m * DDIM + col] = fmaxf(c[ct][r] + bb, 0.0f);
    }
  }
}

// ---------------- kernel 9: out = h @ Wf^T + bf ----------------
__global__ void k_final(const float* __restrict__ hbuf, const float* __restrict__ Wf,
                        const float* __restrict__ bf, float* __restrict__ out, int B) {
  int b = blockIdx.x * blockDim.x + threadIdx.x;
  if (b >= B) return;
  float acc = bf[0];
#pragma unroll 4
  for (int cc = 0; cc < DDIM; ++cc) acc = fmaf(hbuf[(size_t)b * DDIM + cc], Wf[cc], acc);
  out[b] = acc;
}

// ---------------- host launcher ----------------
extern "C" void kernel_launch(void* const* d_in, const int* in_sizes, int n_in,
                              void* d_out, int out_size, void* d_ws, size_t ws_size,
                              hipStream_t stream) {
  const float* z     = (const float*)d_in[0];
  const float* lastu = (const float*)d_in[1];
  const float* t     = (const float*)d_in[2];
  const float* msg   = (const float*)d_in[3];
  const float* timew = (const float*)d_in[4];
  const float* timeb = (const float*)d_in[5];
  const float* Wq  = (const float*)d_in[6];  const float* bq  = (const float*)d_in[7];
  const float* Wk  = (const float*)d_in[8];  const float* bk  = (const float*)d_in[9];
  const float* Wv  = (const float*)d_in[10]; const float* bv  = (const float*)d_in[11];
  const float* We  = (const float*)d_in[12];
  const float* Wsk = (const float*)d_in[13]; const float* bsk = (const float*)d_in[14];
  const float* Wsr = (const float*)d_in[15]; const float* bsr = (const float*)d_in[16];
  const float* Wds = (const float*)d_in[17]; const float* bds = (const float*)d_in[18];
  const float* Wf  = (const float*)d_in[19]; const float* bf  = (const float*)d_in[20];
  const int* n_id       = (const int*)d_in[21];
  const int* srcg       = (const int*)d_in[22];
  const int* dstg       = (const int*)d_in[23];
  const int* eidx       = (const int*)d_in[24];
  const int* numnodes_p = (const int*)d_in[25];
  (void)n_in; (void)out_size;

  const int N = in_sizes[0] / DDIM;  // 100000
  const int E = in_sizes[2];         // 500000
  const int B = in_sizes[22];        // 20000
  const int* e_src = eidx;
  const int* e_dst = eidx + E;

  // carve workspace (256B-aligned regions)
  char* ws = (char*)d_ws;
  size_t off = 0;
  auto carve = [&](size_t bytes) -> char* {
    char* p = ws + off;
    off = (off + bytes + 255) & ~(size_t)255;
    return p;
  };
  int*      assoc = (int*)     carve((size_t)NN_CAP * 4);
  float*    qbuf  = (float*)   carve((size_t)N * DDIM * 4);
  float*    kbuf  = (float*)   carve((size_t)N * DDIM * 4);
  float*    vbuf  = (float*)   carve((size_t)N * DDIM * 4);
  float*    zoutb = (float*)   carve((size_t)N * DDIM * 4);
  float*    eproj = (float*)   carve((size_t)E * DDIM * 4);
  float*    alpha = (float*)   carve((size_t)E * 2 * 4);
  unsigned* amax  = (unsigned*)carve((size_t)N * 2 * 4);
  float*    aden  = (float*)   carve((size_t)N * 2 * 4);
  float*    hbuf  = (float*)   carve((size_t)B * DDIM * 4);
  (void)ws_size;

  k_init <<<2048, 256, 0, stream>>>(assoc, amax, aden, numnodes_p, N);
  k_assoc<<<(N + 255) / 256, 256, 0, stream>>>(assoc, n_id, N);
  {
    int nrt = (N + 15) >> 4;
    int blocks = ((nrt + 7) / 8) * 4;  // 8 row-tiles per block, 4 matrices
    k_qkvs<<<blocks, 256, 0, stream>>>(z, Wq, bq, Wk, bk, Wv, bv, Wsk, bsk,
                                       qbuf, kbuf, vbuf, zoutb, N);
  }
  {
    int waves = (E + 15) >> 4;
    k_eproj<<<(waves + 7) / 8, 256, 0, stream>>>(lastu, t, msg, timew, timeb, We,
                                                 e_src, eproj, E);
  }
  k_alpha <<<(E * 2 + 255) / 256, 256, 0, stream>>>(qbuf, kbuf, eproj, e_src, e_dst,
                                                    alpha, amax, E);
  k_expsum<<<(E * 2 + 255) / 256, 256, 0, stream>>>(e_dst, amax, alpha, aden, E);
  {
    long long tot = (long long)E * DDIM;
    k_agg<<<(int)((tot + 255) / 256), 256, 0, stream>>>(vbuf, eproj, alpha, aden,
                                                        e_src, e_dst, zoutb, E);
  }
  {
    int waves = (B + 15) >> 4;
    k_link<<<(waves + 7) / 8, 256, 0, stream>>>(zoutb, Wsr, bsr, Wds, bds,
                                                assoc, srcg, dstg, hbuf, B);
  }
  k_final<<<(B + 255) / 256, 256, 0, stream>>>(hbuf, Wf, bf, (float*)d_out, B);
}